// SimpleGraphAttentionLayer_2353642078261
// MI455X (gfx1250) — compile-verified
//
#include <hip/hip_runtime.h>
#include <math.h>

#define BS       8
#define NPTS     512
#define IN_DIM   128
#define OUT_DIM  64
#define EDGE_DIM 16
#define ALPHA_STRIDE 516   // 512 + 4 pad -> conflict-free LDS A-fragment reads

typedef float v2f __attribute__((ext_vector_type(2)));
typedef float v8f __attribute__((ext_vector_type(8)));

// ---------------------------------------------------------------------------
// Kernel 1: h[b,i,o] = sum_k x[b,i,k] * W_fc[o,k]   via V_WMMA_F32_16X16X4_F32
// Block: 4 waves (128 thr). Each block: one batch b, 16 rows; wave w owns the
// 16-wide output column tile n0 = 16*w. K loop: 128/4 = 32 WMMA per wave.
// ---------------------------------------------------------------------------
__global__ __launch_bounds__(128)
void gat_h_kernel(const float* __restrict__ x, const float* __restrict__ W,
                  float* __restrict__ h) {
  const int b    = blockIdx.x >> 5;          // 32 row-tiles per batch
  const int i0   = (blockIdx.x & 31) << 4;
  const int wave = threadIdx.x >> 5;
  const int lane = threadIdx.x & 31;
  const int ll   = lane & 15;
  const int hi   = lane >> 4;                // 0: K={k,k+1}, 1: K={k+2,k+3}
  const int n0   = wave << 4;

  const float* xrow = x + (size_t)(b * NPTS + i0 + ll) * IN_DIM;  // A: row M=ll
  const float* wrow = W + (size_t)(n0 + ll) * IN_DIM;             // B: col N=ll (W_fc is [O][I])

  v8f c = {};
  #pragma unroll 4
  for (int k = 0; k < IN_DIM; k += 4) {
    v2f a, bf;
    a.x  = xrow[k + 2 * hi];
    a.y  = xrow[k + 2 * hi + 1];
    bf.x = wrow[k + 2 * hi];
    bf.y = wrow[k + 2 * hi + 1];
    c = __builtin_amdgcn_wmma_f32_16x16x4_f32(false, a, false, bf,
                                              (short)0, c, false, false);
  }
  // C layout: VGPR r -> rows {i0+r (lanes 0-15), i0+r+8 (lanes 16-31)}, col n0+ll
  float* hout = h + (size_t)(b * NPTS + i0 + 8 * hi) * OUT_DIM + n0 + ll;
  #pragma unroll
  for (int r = 0; r < 8; ++r)
    hout[(size_t)r * OUT_DIM] = c[r];
}

// ---------------------------------------------------------------------------
// Kernel 2: s_i[row] = h[row,:]·a_i ; s_j[row] = h[row,:]·a_j   (tiny, 1 MB read)
// ---------------------------------------------------------------------------
__global__ __launch_bounds__(256)
void gat_scores_kernel(const float* __restrict__ h, const float* __restrict__ attn_w,
                       float* __restrict__ s_i, float* __restrict__ s_j) {
  const int row = blockIdx.x * blockDim.x + threadIdx.x;
  if (row >= BS * NPTS) return;
  const float* hr = h + (size_t)row * OUT_DIM;
  float si = 0.f, sj = 0.f;
  #pragma unroll 8
  for (int o = 0; o < OUT_DIM; ++o) {
    const float v = hr[o];
    si += v * attn_w[o];
    sj += v * attn_w[OUT_DIM + o];
  }
  s_i[row] = si;
  s_j[row] = sj;
}

// ---------------------------------------------------------------------------
// Kernel 3 (bandwidth-dominant): per block = (b, 16 rows i), 16 waves.
// Phase 1: wave w streams edge_feats[b, i0+w, :, :] once (512 x 64B, lanes cover
//          32 consecutive j -> 2KB contiguous per wave-load), computes logits,
//          wave32 shfl-reduce max/sum softmax, alpha -> LDS (stride 516).
// Phase 2: waves 0-3 run h' = alpha @ h with V_WMMA_F32_16X16X4_F32 (K=512).
// ---------------------------------------------------------------------------
__global__ __launch_bounds__(512)
void gat_attn_kernel(const float* __restrict__ ef, const float* __restrict__ attn_w,
                     const float* __restrict__ h, const float* __restrict__ s_i,
                     const float* __restrict__ s_j, float* __restrict__ out) {
  __shared__ float alpha_s[16 * ALPHA_STRIDE];   // 33 KB of 320 KB WGP LDS

  const int b    = blockIdx.x >> 5;
  const int i0   = (blockIdx.x & 31) << 4;
  const int wave = threadIdx.x >> 5;
  const int lane = threadIdx.x & 31;

  // ---- Phase 1: fused s_e + bias + softmax for row i = i0 + wave ----
  const int i = i0 + wave;
  float ae[EDGE_DIM];
  #pragma unroll
  for (int q = 0; q < EDGE_DIM; ++q) ae[q] = attn_w[2 * OUT_DIM + q];  // uniform -> s_load

  const float  sP    = s_i[b * NPTS + i];
  const float* efrow = ef + (size_t)(b * NPTS + i) * NPTS * EDGE_DIM;

  float e[16];
  float mx = -INFINITY;
  #pragma unroll
  for (int t = 0; t < 16; ++t) {
    const int j = t * 32 + lane;
    const float4* p = (const float4*)(efrow + (size_t)j * EDGE_DIM);
    const float4 f0 = p[0], f1 = p[1], f2 = p[2], f3 = p[3];  // 4x global_load_b128
    const float d =
        f0.x * ae[0]  + f0.y * ae[1]  + f0.z * ae[2]  + f0.w * ae[3]  +
        f1.x * ae[4]  + f1.y * ae[5]  + f1.z * ae[6]  + f1.w * ae[7]  +
        f2.x * ae[8]  + f2.y * ae[9]  + f2.z * ae[10] + f2.w * ae[11] +
        f3.x * ae[12] + f3.y * ae[13] + f3.z * ae[14] + f3.w * ae[15];
    const float ev = sP + s_j[b * NPTS + j] + d;
    e[t] = ev;
    mx = fmaxf(mx, ev);
  }
  for (int off = 16; off; off >>= 1) mx = fmaxf(mx, __shfl_xor(mx, off, 32));

  float sum = 0.f;
  #pragma unroll
  for (int t = 0; t < 16; ++t) { e[t] = __expf(e[t] - mx); sum += e[t]; }
  for (int off = 16; off; off >>= 1) sum += __shfl_xor(sum, off, 32);

  const float inv = 1.0f / sum;
  #pragma unroll
  for (int t = 0; t < 16; ++t)
    alpha_s[wave * ALPHA_STRIDE + t * 32 + lane] = e[t] * inv;

  __syncthreads();

  // ---- Phase 2: h'[i0:i0+16, :] = alpha(16x512) @ h_b(512x64) ----
  if (wave < 4) {
    const int ll = lane & 15;
    const int hi = lane >> 4;
    const int n0 = wave << 4;
    const float* hb = h + (size_t)b * NPTS * OUT_DIM + n0 + ll;  // B col N=n0+ll

    v8f c = {};
    for (int k = 0; k < NPTS; k += 4) {
      v2f a, bf;
      a.x  = alpha_s[ll * ALPHA_STRIDE + k + 2 * hi];
      a.y  = alpha_s[ll * ALPHA_STRIDE + k + 2 * hi + 1];
      bf.x = hb[(size_t)(k + 2 * hi) * OUT_DIM];       // h hot in L2 (1 MB / 192 MB)
      bf.y = hb[(size_t)(k + 2 * hi + 1) * OUT_DIM];
      c = __builtin_amdgcn_wmma_f32_16x16x4_f32(false, a, false, bf,
                                                (short)0, c, false, false);
    }
    float* o = out + (size_t)(b * NPTS + i0 + 8 * hi) * OUT_DIM + n0 + ll;
    #pragma unroll
    for (int r = 0; r < 8; ++r)
      o[(size_t)r * OUT_DIM] = c[r];
  }
}

// ---------------------------------------------------------------------------
extern "C" void kernel_launch(void* const* d_in, const int* in_sizes, int n_in,
                              void* d_out, int out_size, void* d_ws, size_t ws_size,
                              hipStream_t stream) {
  (void)in_sizes; (void)n_in; (void)out_size; (void)ws_size;
  const float* x   = (const float*)d_in[0];   // (8,512,128)
  const float* ef  = (const float*)d_in[1];   // (8,512,512,16)
  const float* W   = (const float*)d_in[2];   // (64,128)
  const float* aw  = (const float*)d_in[3];   // (144,)
  float* out = (float*)d_out;                 // (8,512,64)

  float* h   = (float*)d_ws;                          // 8*512*64 f32 = 1 MB
  float* s_i = h + (size_t)BS * NPTS * OUT_DIM;       // 8*512 f32
  float* s_j = s_i + (size_t)BS * NPTS;               // 8*512 f32

  gat_h_kernel<<<BS * 32, 128, 0, stream>>>(x, W, h);
  gat_scores_kernel<<<(BS * NPTS + 255) / 256, 256, 0, stream>>>(h, aw, s_i, s_j);
  gat_attn_kernel<<<BS * 32, 512, 0, stream>>>(ef, aw, h, s_i, s_j, out);
}